// SGFormer_89455578841349
// MI455X (gfx1250) — compile-verified
//
#include <hip/hip_runtime.h>
#include <hip/hip_bf16.h>
#include <math.h>

// ---------------- problem constants (match reference) ----------------
#define TOTN   16384      // total nodes (B*N)
#define DM     256        // model dim
#define NEDGE  262144     // edges
#define NGRAPH 64         // graphs
#define NPG    256        // nodes per graph
#define NHEAD  4
#define HDIM   64

typedef float f32x2 __attribute__((ext_vector_type(2)));
typedef float f32x8 __attribute__((ext_vector_type(8)));

typedef __attribute__((address_space(1))) int g_i32;   // global addrspace
typedef __attribute__((address_space(3))) int l_i32;   // LDS addrspace

static __device__ __forceinline__ f32x8 wmma_f32_16x16x4(f32x2 a, f32x2 b, f32x8 c) {
    // (neg_a, A, neg_b, B, c_mod, C, reuse_a, reuse_b) -- 8-arg pattern for 16x16x{4,32}
    return __builtin_amdgcn_wmma_f32_16x16x4_f32(false, a, false, b, (short)0, c, false, false);
}

// gfx1250 async global->LDS path (ASYNCcnt), guarded so compile cannot regress.
#if defined(__has_builtin)
#if __has_builtin(__builtin_amdgcn_global_load_async_to_lds_b32) && \
    __has_builtin(__builtin_amdgcn_s_wait_asynccnt)
#define USE_ASYNC_LDS 1
#endif
#endif

// =====================================================================
// GEMM: C[M,256] = X[M,256] @ W[256,256] + bias, fp32 WMMA 16x16x4.
// Block = 256 thr = 8 waves. Wave => 16 rows x 64 cols (4 C tiles).
// Block tile = 128 rows x 64 cols. grid = (M/128, 4).
// The 64-col W panel (64KB) is staged once per block into LDS as
// K-pair float2s: wp[p*64+c] = {W[2p][c0+c], W[2p+1][c0+c]}, so each
// B fragment is one ds_load_b64.
// =====================================================================
__global__ __launch_bounds__(256) void gemm256_bias(
    const float* __restrict__ X, const float* __restrict__ W,
    const float* __restrict__ bias, float* __restrict__ C, int M)
{
    __shared__ f32x2 wp[128 * 64];   // 64 KB

    const int lane = threadIdx.x & 31;
    const int wave = threadIdx.x >> 5;
    const int hl   = lane >> 4;      // lane half: selects K pair
    const int l16  = lane & 15;
    const int r0   = blockIdx.x * 128 + wave * 16;
    const int c0   = blockIdx.y * 64;

    // ---- stage W panel into LDS (pair-transposed) ----
    for (int idx = threadIdx.x; idx < 128 * 64; idx += 256) {
        const int p = idx >> 6;
        const int c = idx & 63;
        const float* g0 = W + (size_t)(2 * p) * DM + c0 + c;
#if USE_ASYNC_LDS
        float* l0 = reinterpret_cast<float*>(&wp[idx]);
        __builtin_amdgcn_global_load_async_to_lds_b32(
            (g_i32*)(g0), (l_i32*)(l0), 0, 0);
        __builtin_amdgcn_global_load_async_to_lds_b32(
            (g_i32*)(g0 + DM), (l_i32*)(l0 + 1), 0, 0);
#else
        f32x2 t; t.x = g0[0]; t.y = g0[DM];
        wp[idx] = t;
#endif
    }
#if USE_ASYNC_LDS
    __builtin_amdgcn_s_wait_asynccnt(0);
#endif
    __syncthreads();

    f32x8 acc0 = {0,0,0,0,0,0,0,0}, acc1 = acc0, acc2 = acc0, acc3 = acc0;
    const float* __restrict__ xrow = X + (size_t)(r0 + l16) * DM;

    #pragma unroll 4
    for (int kk = 0; kk < DM; kk += 4) {
        const int p  = (kk >> 1) + hl;           // float2 pair row in LDS
        const int k0 = kk + 2 * hl;
        const f32x2 a  = *reinterpret_cast<const f32x2*>(xrow + k0);  // b64
        const f32x2 b0 = wp[p * 64 + l16];        // ds_load_b64 each
        const f32x2 b1 = wp[p * 64 + 16 + l16];
        const f32x2 b2 = wp[p * 64 + 32 + l16];
        const f32x2 b3 = wp[p * 64 + 48 + l16];
        acc0 = wmma_f32_16x16x4(a, b0, acc0);
        acc1 = wmma_f32_16x16x4(a, b1, acc1);
        acc2 = wmma_f32_16x16x4(a, b2, acc2);
        acc3 = wmma_f32_16x16x4(a, b3, acc3);
    }
    const float bi0 = bias[c0 + l16];
    const float bi1 = bias[c0 + l16 + 16];
    const float bi2 = bias[c0 + l16 + 32];
    const float bi3 = bias[c0 + l16 + 48];
    #pragma unroll
    for (int vr = 0; vr < 8; ++vr) {
        const int row = r0 + vr + 8 * hl;
        float* crow = C + (size_t)row * DM + c0 + l16;
        crow[0]  = acc0[vr] + bi0;
        crow[16] = acc1[vr] + bi1;
        crow[32] = acc2[vr] + bi2;
        crow[48] = acc3[vr] + bi3;
    }
}

// =====================================================================
// CSR build
// =====================================================================
__global__ void zero_i32(int* __restrict__ p, int n) {
    int i = blockIdx.x * blockDim.x + threadIdx.x;
    if (i < n) p[i] = 0;
}

__global__ void edge_hist(const int* __restrict__ dst, int* __restrict__ cnt, int e) {
    int i = blockIdx.x * blockDim.x + threadIdx.x;
    if (i < e) atomicAdd(&cnt[dst[i]], 1);
}

__global__ __launch_bounds__(256) void scan_offsets(
    const int* __restrict__ cnt, int* __restrict__ off, int n)
{
    __shared__ int sums[256];
    const int t = threadIdx.x;
    const int chunk = n / 256;        // 64
    const int base = t * chunk;
    int s = 0;
    for (int j = 0; j < chunk; ++j) s += cnt[base + j];
    sums[t] = s;
    __syncthreads();
    for (int d = 1; d < 256; d <<= 1) {
        int v = (t >= d) ? sums[t - d] : 0;
        __syncthreads();
        sums[t] += v;
        __syncthreads();
    }
    int run = (t == 0) ? 0 : sums[t - 1];   // exclusive prefix
    for (int j = 0; j < chunk; ++j) { off[base + j] = run; run += cnt[base + j]; }
    if (t == 255) off[n] = run;
}

__global__ void edge_scatter(const int* __restrict__ dst, const int* __restrict__ off,
                             int* __restrict__ fill, int* __restrict__ csr, int e)
{
    int i = blockIdx.x * blockDim.x + threadIdx.x;
    if (i < e) {
        int d = dst[i];
        int p = atomicAdd(&fill[d], 1);
        csr[off[d] + p] = i;
    }
}

// =====================================================================
// Local branch: per-destination-node edge softmax + aggregation.
// One wave per node; each lane owns 8 of the 256 feature dims.
// =====================================================================
__global__ __launch_bounds__(256) void local_attn(
    const float* __restrict__ q, const float* __restrict__ k,
    const float* __restrict__ v, const int* __restrict__ src,
    const int* __restrict__ off, const int* __restrict__ csr,
    float* __restrict__ logits, float* __restrict__ agg)
{
    const int lane = threadIdx.x & 31;
    const int wave = threadIdx.x >> 5;
    const int node = blockIdx.x * 8 + wave;
    const int beg = off[node], end = off[node + 1];
    const size_t fbase = (size_t)node * DM + lane * 8;

    float4 q0 = *reinterpret_cast<const float4*>(q + fbase);
    float4 q1 = *reinterpret_cast<const float4*>(q + fbase + 4);

    // pass 1: logits + running max
    float m = -3.4e38f;
    for (int p = beg; p < end; ++p) {
        const int e = csr[p];
        const int s = src[e];
        const float* kp = k + (size_t)s * DM + lane * 8;
        float4 k0 = *reinterpret_cast<const float4*>(kp);
        float4 k1 = *reinterpret_cast<const float4*>(kp + 4);
        float part = q0.x * k0.x + q0.y * k0.y + q0.z * k0.z + q0.w * k0.w
                   + q1.x * k1.x + q1.y * k1.y + q1.z * k1.z + q1.w * k1.w;
        #pragma unroll
        for (int o = 16; o >= 1; o >>= 1) part += __shfl_xor(part, o);
        const float lg = part * (1.0f / 16.0f);   // 1/sqrt(256)
        if (lane == 0) logits[p] = lg;
        m = fmaxf(m, lg);
    }

    // pass 2: exp-sum + weighted aggregation of v
    float acc[8];
    #pragma unroll
    for (int j = 0; j < 8; ++j) acc[j] = 0.f;
    float den = 0.f;
    for (int p = beg; p < end; ++p) {
        const int e = csr[p];
        const int s = src[e];
        const float ex = __expf(logits[p] - m);
        den += ex;
        const float* vp = v + (size_t)s * DM + lane * 8;
        float4 v0 = *reinterpret_cast<const float4*>(vp);
        float4 v1 = *reinterpret_cast<const float4*>(vp + 4);
        acc[0] += ex * v0.x; acc[1] += ex * v0.y; acc[2] += ex * v0.z; acc[3] += ex * v0.w;
        acc[4] += ex * v1.x; acc[5] += ex * v1.y; acc[6] += ex * v1.z; acc[7] += ex * v1.w;
    }
    const float inv = 1.0f / (den + 1e-16f);
    #pragma unroll
    for (int j = 0; j < 8; ++j) agg[fbase + j] = acc[j] * inv;
}

// =====================================================================
// Global branch: dense per-graph multihead attention with WMMA.
// One wave per (graph b, head h, 16 query rows). Mask is all-true
// (every graph has exactly NPG nodes), so plain softmax.
// =====================================================================
__global__ __launch_bounds__(32) void global_attn(
    const float* __restrict__ gq, const float* __restrict__ gk,
    const float* __restrict__ gv, float* __restrict__ gatt)
{
    __shared__ float S[16][258];   // padded: stride 258 de-conflicts banks
    const int lane = threadIdx.x;
    const int hl   = lane >> 4;
    const int l16  = lane & 15;
    const int q0   = blockIdx.x * 16;
    const int h    = blockIdx.y;
    const int b    = blockIdx.z;

    const float* __restrict__ Qb = gq + ((size_t)b * NPG + q0) * DM + h * HDIM;
    const float* __restrict__ Kb = gk + (size_t)b * NPG * DM + h * HDIM;
    const float* __restrict__ Vb = gv + (size_t)b * NPG * DM + h * HDIM;

    // ---- S = Q * K^T  (16 x 256) ----
    f32x8 acc[16];
    #pragma unroll
    for (int t = 0; t < 16; ++t) acc[t] = (f32x8){0,0,0,0,0,0,0,0};

    for (int kk = 0; kk < HDIM; kk += 4) {
        const int k0 = kk + 2 * hl;
        f32x2 a = *reinterpret_cast<const f32x2*>(Qb + (size_t)l16 * DM + k0);
        #pragma unroll
        for (int t = 0; t < 16; ++t) {
            const float* kp = Kb + (size_t)(t * 16 + l16) * DM + k0;  // B[k][n] = K[n][k]
            f32x2 bf = *reinterpret_cast<const f32x2*>(kp);
            acc[t] = wmma_f32_16x16x4(a, bf, acc[t]);
        }
    }
    #pragma unroll
    for (int t = 0; t < 16; ++t)
        #pragma unroll
        for (int vr = 0; vr < 8; ++vr)
            S[vr + 8 * hl][t * 16 + l16] = acc[t][vr] * 0.125f;   // 1/sqrt(64)
    __syncthreads();

    // ---- row softmax: 2 lanes per row, 128 cols each ----
    {
        const int row = lane >> 1;
        const int cb  = (lane & 1) * 128;
        float mx = -3.4e38f;
        for (int c = 0; c < 128; ++c) mx = fmaxf(mx, S[row][cb + c]);
        mx = fmaxf(mx, __shfl_xor(mx, 1));
        float sum = 0.f;
        for (int c = 0; c < 128; ++c) {
            float e = __expf(S[row][cb + c] - mx);
            S[row][cb + c] = e;
            sum += e;
        }
        sum += __shfl_xor(sum, 1);
        const float inv = 1.0f / sum;
        for (int c = 0; c < 128; ++c) S[row][cb + c] *= inv;
    }
    __syncthreads();

    // ---- O = P * V  (16 x 64) ----
    f32x8 o[4];
    #pragma unroll
    for (int t = 0; t < 4; ++t) o[t] = (f32x8){0,0,0,0,0,0,0,0};

    for (int kk = 0; kk < NPG; kk += 4) {
        const int k0 = kk + 2 * hl;
        f32x2 a; a.x = S[l16][k0]; a.y = S[l16][k0 + 1];
        #pragma unroll
        for (int t = 0; t < 4; ++t) {
            const float* vp = Vb + (size_t)k0 * DM + t * 16 + l16;
            f32x2 bf; bf.x = vp[0]; bf.y = vp[DM];
            o[t] = wmma_f32_16x16x4(a, bf, o[t]);
        }
    }
    #pragma unroll
    for (int t = 0; t < 4; ++t)
        #pragma unroll
        for (int vr = 0; vr < 8; ++vr)
            gatt[((size_t)b * NPG + q0 + vr + 8 * hl) * DM + h * HDIM + t * 16 + l16] = o[t][vr];
}

// =====================================================================
// Fusion + LayerNorm: one wave per node, lane owns 8 dims.
// out = LN( wl*(agg + sloc) + wg*(gproj + x) + x ) * g + b
// =====================================================================
__global__ __launch_bounds__(256) void fuse_ln(
    const float* __restrict__ x, const float* __restrict__ agg,
    const float* __restrict__ sloc, const float* __restrict__ gproj,
    const float* __restrict__ wlp, const float* __restrict__ wgp,
    const float* __restrict__ lng, const float* __restrict__ lnb,
    float* __restrict__ out)
{
    const int lane = threadIdx.x & 31;
    const int wave = threadIdx.x >> 5;
    const int node = blockIdx.x * 8 + wave;
    const float wl = wlp[0], wg = wgp[0];
    const size_t base = (size_t)node * DM + lane * 8;

    float t[8];
    float s = 0.f;
    #pragma unroll
    for (int j = 0; j < 8; ++j) {
        const float xx = x[base + j];
        const float f = wl * (agg[base + j] + sloc[base + j]) + wg * (gproj[base + j] + xx) + xx;
        t[j] = f;
        s += f;
    }
    #pragma unroll
    for (int o = 16; o >= 1; o >>= 1) s += __shfl_xor(s, o);
    const float mu = s * (1.0f / DM);
    float vs = 0.f;
    #pragma unroll
    for (int j = 0; j < 8; ++j) { const float d = t[j] - mu; vs += d * d; }
    #pragma unroll
    for (int o = 16; o >= 1; o >>= 1) vs += __shfl_xor(vs, o);
    const float inv = rsqrtf(vs * (1.0f / DM) + 1e-5f);
    #pragma unroll
    for (int j = 0; j < 8; ++j) {
        const int col = lane * 8 + j;
        out[base + j] = (t[j] - mu) * inv * lng[col] + lnb[col];
    }
}

// =====================================================================
extern "C" void kernel_launch(void* const* d_in, const int* in_sizes, int n_in,
                              void* d_out, int out_size, void* d_ws, size_t ws_size,
                              hipStream_t stream)
{
    (void)in_sizes; (void)n_in; (void)out_size; (void)ws_size;

    const float* x    = (const float*)d_in[0];
    const int*   ei   = (const int*)d_in[1];          // [2, E] int32
    const int*   esrc = ei;
    const int*   edst = ei + NEDGE;
    // d_in[2] = batch (implicit: i / NPG, unused)
    const float* lq_w = (const float*)d_in[3];  const float* lq_b = (const float*)d_in[4];
    const float* lk_w = (const float*)d_in[5];  const float* lk_b = (const float*)d_in[6];
    const float* lv_w = (const float*)d_in[7];  const float* lv_b = (const float*)d_in[8];
    const float* ls_w = (const float*)d_in[9];  const float* ls_b = (const float*)d_in[10];
    const float* gq_w = (const float*)d_in[11]; const float* gq_b = (const float*)d_in[12];
    const float* gk_w = (const float*)d_in[13]; const float* gk_b = (const float*)d_in[14];
    const float* gv_w = (const float*)d_in[15]; const float* gv_b = (const float*)d_in[16];
    const float* go_w = (const float*)d_in[17]; const float* go_b = (const float*)d_in[18];
    const float* wlp  = (const float*)d_in[19];
    const float* wgp  = (const float*)d_in[20];
    const float* lng  = (const float*)d_in[21];
    const float* lnb  = (const float*)d_in[22];
    float* out = (float*)d_out;

    // ---- workspace layout ----
    float* ws = (float*)d_ws;
    const size_t P = (size_t)TOTN * DM;   // 4,194,304 floats
    float* q     = ws + 0 * P;
    float* kbuf  = ws + 1 * P;
    float* vbuf  = ws + 2 * P;
    float* sloc  = ws + 3 * P;
    float* gq    = ws + 4 * P;
    float* gk    = ws + 5 * P;
    float* gv    = ws + 6 * P;
    float* gatt  = ws + 7 * P;
    float* agg   = ws + 8 * P;
    float* logits= ws + 9 * P;            // E floats
    int*   ib    = (int*)(ws + 9 * P + NEDGE);
    int*   cnt   = ib;                    // TOTN
    int*   off   = ib + TOTN;             // TOTN+1
    int*   fill  = ib + 2 * TOTN + 1;     // TOTN
    int*   csr   = ib + 3 * TOTN + 1;     // E
    float* gproj = q;                     // reuse q after local_attn

    const dim3 gemm_grid(TOTN / 128, 4);
    const dim3 gemm_blk(256);

    // ---- 7 input projections ----
    gemm256_bias<<<gemm_grid, gemm_blk, 0, stream>>>(x, lq_w, lq_b, q,    TOTN);
    gemm256_bias<<<gemm_grid, gemm_blk, 0, stream>>>(x, lk_w, lk_b, kbuf, TOTN);
    gemm256_bias<<<gemm_grid, gemm_blk, 0, stream>>>(x, lv_w, lv_b, vbuf, TOTN);
    gemm256_bias<<<gemm_grid, gemm_blk, 0, stream>>>(x, ls_w, ls_b, sloc, TOTN);
    gemm256_bias<<<gemm_grid, gemm_blk, 0, stream>>>(x, gq_w, gq_b, gq,   TOTN);
    gemm256_bias<<<gemm_grid, gemm_blk, 0, stream>>>(x, gk_w, gk_b, gk,   TOTN);
    gemm256_bias<<<gemm_grid, gemm_blk, 0, stream>>>(x, gv_w, gv_b, gv,   TOTN);

    // ---- CSR build over dst ----
    zero_i32<<<TOTN / 256, 256, 0, stream>>>(cnt, TOTN);
    zero_i32<<<TOTN / 256, 256, 0, stream>>>(fill, TOTN);
    edge_hist<<<NEDGE / 256, 256, 0, stream>>>(edst, cnt, NEDGE);
    scan_offsets<<<1, 256, 0, stream>>>(cnt, off, TOTN);
    edge_scatter<<<NEDGE / 256, 256, 0, stream>>>(edst, off, fill, csr, NEDGE);

    // ---- local attention ----
    local_attn<<<TOTN / 8, 256, 0, stream>>>(q, kbuf, vbuf, esrc, off, csr, logits, agg);

    // ---- global attention ----
    global_attn<<<dim3(NPG / 16, NHEAD, NGRAPH), 32, 0, stream>>>(gq, gk, gv, gatt);

    // ---- output projection of global branch (reuses q buffer) ----
    gemm256_bias<<<gemm_grid, gemm_blk, 0, stream>>>(gatt, go_w, go_b, gproj, TOTN);

    // ---- fuse + residual + LayerNorm ----
    fuse_ln<<<TOTN / 8, 256, 0, stream>>>(x, agg, sloc, gproj, wlp, wgp, lng, lnb, out);
}